// GCN_31920196944512
// MI455X (gfx1250) — compile-verified
//
#include <hip/hip_runtime.h>
#include <hip/hip_bf16.h>

// ---------------------------------------------------------------------------
// GCN point-cloud classifier for MI455X (gfx1250, wave32).
// A_hat row sums are always K+1=33 (kNN always contains self), so
// dinv == 1/sqrt(33) and A_norm @ H == (sum_k H[idx[k]] + H[i]) / 33.
// Heavy GEMMs use fp32 WMMA 16x16x4 with LDS double-buffering; tiles filled
// with GLOBAL_LOAD_ASYNC_TO_LDS_B128/B32 (ASYNCcnt). B is stored k-pair
// interleaved in LDS so each WMMA B-fragment is one aligned ds_load_b64.
// 512->1024 layer fuses BN+ReLU+max-pool (atomicMax on non-negative floats).
// ---------------------------------------------------------------------------

#define B_       16
#define N_       2048
#define KNN_     32
#define NPTS_    (B_ * N_)      // 32768

typedef __attribute__((ext_vector_type(2))) float v2f;
typedef __attribute__((ext_vector_type(8))) float v8f;

#if __has_builtin(__builtin_amdgcn_global_load_async_to_lds_b128) && \
    __has_builtin(__builtin_amdgcn_s_wait_asynccnt)
#define USE_ASYNC_LDS 1
#else
#define USE_ASYNC_LDS 0
#endif

#if __has_builtin(__builtin_amdgcn_global_load_async_to_lds_b32) && \
    __has_builtin(__builtin_amdgcn_s_wait_asynccnt)
#define USE_ASYNC_LDS32 1
#else
#define USE_ASYNC_LDS32 0
#endif

#if USE_ASYNC_LDS
// param types per hipcc diagnostic: 16-byte int vector, AS1 / AS3 pointers
typedef int vec4i_ __attribute__((vector_size(16)));
typedef __attribute__((address_space(1))) vec4i_* g4ptr_;
typedef __attribute__((address_space(3))) vec4i_* l4ptr_;
#endif
#if USE_ASYNC_LDS32
typedef __attribute__((address_space(1))) int* g1ptr_;
typedef __attribute__((address_space(3))) int* l1ptr_;
#endif

// 16-byte global -> LDS copy (async when the gfx1250 builtin is available)
__device__ __forceinline__ void copy16_to_lds(const float* __restrict__ g,
                                              float* __restrict__ l) {
#if USE_ASYNC_LDS
    __builtin_amdgcn_global_load_async_to_lds_b128((g4ptr_)g, (l4ptr_)l, 0, 0);
#else
    *(float4*)l = *(const float4*)g;    // global_load_b128 + ds_store_b128
#endif
}

// 4-byte global -> LDS copy (for the interleaved B layout)
__device__ __forceinline__ void copy4_to_lds(const float* __restrict__ g,
                                             float* __restrict__ l) {
#if USE_ASYNC_LDS32
    __builtin_amdgcn_global_load_async_to_lds_b32((g1ptr_)g, (l1ptr_)l, 0, 0);
#else
    *l = *g;                            // global_load_b32 + ds_store_b32
#endif
}

__device__ __forceinline__ void lds_copy_wait() {
#if USE_ASYNC_LDS || USE_ASYNC_LDS32
    __builtin_amdgcn_s_wait_asynccnt(0);
#endif
}

// ---------------------------------------------------------------- kNN top-K
__global__ __launch_bounds__(256)
void knn_kernel(const float* __restrict__ x, int* __restrict__ idx) {
    const int p = blockIdx.x;            // global point id
    const int b = p >> 11;               // /2048
    const int i = p & (N_ - 1);
    const float* xb = x + (size_t)b * N_ * 3;

    __shared__ float d2s[N_];
    __shared__ float rmin[256];
    __shared__ int   rarg[256];

    const float xi0 = xb[i * 3 + 0];
    const float xi1 = xb[i * 3 + 1];
    const float xi2 = xb[i * 3 + 2];
    const int tid = threadIdx.x;

    for (int j = tid; j < N_; j += 256) {
        const float dx = xb[j * 3 + 0] - xi0;
        const float dy = xb[j * 3 + 1] - xi1;
        const float dz = xb[j * 3 + 2] - xi2;
        d2s[j] = dx * dx + dy * dy + dz * dz;
    }
    __syncthreads();

    for (int k = 0; k < KNN_; ++k) {
        float best = 3.402823466e+38f;
        int   bi   = 0x7fffffff;
        for (int j = tid; j < N_; j += 256) {
            const float v = d2s[j];
            if (v < best || (v == best && j < bi)) { best = v; bi = j; }
        }
        rmin[tid] = best; rarg[tid] = bi;
        __syncthreads();
        for (int s = 128; s > 0; s >>= 1) {
            if (tid < s) {
                if (rmin[tid + s] < rmin[tid] ||
                    (rmin[tid + s] == rmin[tid] && rarg[tid + s] < rarg[tid])) {
                    rmin[tid] = rmin[tid + s];
                    rarg[tid] = rarg[tid + s];
                }
            }
            __syncthreads();
        }
        if (tid == 0) {
            idx[(size_t)p * KNN_ + k] = rarg[0];
            d2s[rarg[0]] = 3.402823466e+38f;   // remove winner
        }
        __syncthreads();
    }
}

// --------------------------------------------------- x(B,N,3) @ W(3,64) + b
__global__ __launch_bounds__(256)
void gemm_g1_kernel(const float* __restrict__ x, const float* __restrict__ W,
                    const float* __restrict__ bias, float* __restrict__ out) {
    const int t = blockIdx.x * 256 + threadIdx.x;   // p*64 + f
    const int p = t >> 6;
    const int f = t & 63;
    const float* xp = x + (size_t)p * 3;
    out[t] = xp[0] * W[0 * 64 + f] + xp[1] * W[1 * 64 + f] +
             xp[2] * W[2 * 64 + f] + bias[f];
}

// ---------------------------- out[i] = relu((H[i] + sum_k H[idx[i,k]]) / 33)
template <int F>
__global__ void agg_relu_kernel(const float* __restrict__ h,
                                const int* __restrict__ idx,
                                float* __restrict__ out) {
    const int p = blockIdx.x;
    const int b = p >> 11;
    const int f = threadIdx.x;                      // F threads per block
    const float* hb = h + (size_t)b * N_ * F;       // batch base
    const int*   ip = idx + (size_t)p * KNN_;
    float s = h[(size_t)p * F + f];                 // self (identity term)
    #pragma unroll 4
    for (int k = 0; k < KNN_; ++k)
        s += hb[(size_t)ip[k] * F + f];
    out[(size_t)p * F + f] = fmaxf(s * (1.0f / 33.0f), 0.0f);
}

// ------------------------------------------------------------- WMMA GEMM f32
// Block = 128 threads = 4 waves; block tile 128(M) x 64(N), K chunk = 16.
// A staged row-major (b128 async fills); B staged k-pair interleaved
// (b32 async fills) so each B fragment is one aligned ds_load_b64.
// Each wave owns a 32x64 sub-tile = 2x4 WMMA 16x16x4 accumulators.
// Epilogue: + bias, optional eval-BN (scale = g*rsqrt(1+1e-5), shift = be),
// optional ReLU, optional fused max-pool over the 2048 points of each batch.
#define BK_      16
#define APAD_    4       // A LDS row stride 20 words (conflict-free frag reads)
#define BPAD_    2       // B pair-row stride 130 words
template <bool HAS_BN, bool RELU, bool POOL>
__global__ __launch_bounds__(128)
void wmma_gemm_kernel(const float* __restrict__ A, const float* __restrict__ W,
                      const float* __restrict__ bias,
                      const float* __restrict__ gamma,
                      const float* __restrict__ beta,
                      float* __restrict__ out, unsigned int* __restrict__ pool,
                      int K, int N) {
    __shared__ float As[2][128][BK_ + APAD_];
    // Bs[buf][kp][n*2 + parity] = W[kc + 2*kp + parity][col0 + n]
    __shared__ float Bs[2][BK_ / 2][64 * 2 + BPAD_];

    const int tid   = threadIdx.x;
    const int wave  = tid >> 5;
    const int lane  = tid & 31;
    const int lhalf = lane >> 4;        // 0: lanes 0-15, 1: lanes 16-31
    const int l15   = lane & 15;
    const int rowB  = blockIdx.x * 128;             // block row base
    const int col0  = blockIdx.y * 64;

    v8f c[2][4] = {};   // 2 M-tiles x 4 N-tiles per wave

    // cooperative tile fill: A 128x16 (4 x b128/thread), B 16x64 (8 x b32/thread)
    const int arow = tid >> 2;          // 0..31
    const int aseg = (tid & 3) * 4;     // 0,4,8,12
    auto load_chunk = [&](int ch, int bsel) {
        const int kc = ch * BK_;
        #pragma unroll
        for (int rr = 0; rr < 4; ++rr) {
            const int row = rr * 32 + arow;
            copy16_to_lds(A + (size_t)(rowB + row) * K + kc + aseg,
                          &As[bsel][row][aseg]);
        }
        #pragma unroll
        for (int i = 0; i < 8; ++i) {
            const int e = tid + i * 128;            // 0..1023 elements
            const int k = e >> 6;                   // 0..15
            const int n = e & 63;                   // 0..63 (coalesced per lane)
            copy4_to_lds(W + (size_t)(kc + k) * N + col0 + n,
                         &Bs[bsel][k >> 1][n * 2 + (k & 1)]);
        }
    };

    const int nch = K / BK_;
    load_chunk(0, 0);
    lds_copy_wait();
    __syncthreads();

    for (int ch = 0; ch < nch; ++ch) {
        const int cur = ch & 1;
        if (ch + 1 < nch) load_chunk(ch + 1, cur ^ 1);

        #pragma unroll
        for (int kk = 0; kk < BK_; kk += 4) {
            // A frag: lanes 0-15 -> (k,k+1) of row M=l15; lanes 16-31 -> (k+2,k+3)
            v2f a[2];
            #pragma unroll
            for (int mt = 0; mt < 2; ++mt)
                a[mt] = *(const v2f*)&As[cur][wave * 32 + mt * 16 + l15]
                                        [kk + lhalf * 2];
            // B frag: one aligned b64 from the interleaved pair layout
            v2f bf[4];
            #pragma unroll
            for (int nt = 0; nt < 4; ++nt)
                bf[nt] = *(const v2f*)&Bs[cur][(kk >> 1) + lhalf]
                                         [(nt * 16 + l15) * 2];
            #pragma unroll
            for (int mt = 0; mt < 2; ++mt)
                #pragma unroll
                for (int nt = 0; nt < 4; ++nt)
                    c[mt][nt] = __builtin_amdgcn_wmma_f32_16x16x4_f32(
                        false, a[mt], false, bf[nt], (short)0, c[mt][nt],
                        false, false);
        }
        lds_copy_wait();     // our async fills for buffer cur^1 landed
        __syncthreads();     // everyone done reading cur before next overwrite
    }

    // Epilogue. C layout: VGPR r, lane l -> m = r + (l>=16 ? 8 : 0), n = l&15.
    #pragma unroll
    for (int nt = 0; nt < 4; ++nt) {
        const int n = col0 + nt * 16 + l15;
        const float bs = bias[n];
        float sc = 1.0f, sh = 0.0f;
        if (HAS_BN) { sc = gamma[n] * rsqrtf(1.0f + 1e-5f); sh = beta[n]; }
        #pragma unroll
        for (int mt = 0; mt < 2; ++mt) {
            #pragma unroll
            for (int r = 0; r < 8; ++r) {
                const int m = rowB + wave * 32 + mt * 16 + lhalf * 8 + r;
                float v = c[mt][nt][r] + bs;
                if (HAS_BN) v = v * sc + sh;
                if (RELU)   v = fmaxf(v, 0.0f);
                if (POOL) {
                    // max over points of batch (m>>11); v >= 0 after relu, so
                    // uint bit-pattern compare == float compare
                    atomicMax(&pool[(size_t)(m >> 11) * N + n],
                              __float_as_uint(v));
                } else {
                    out[(size_t)m * N + n] = v;
                }
            }
        }
    }
}

// ---------------------------------------------------------------- zero init
__global__ void zero_kernel(unsigned int* __restrict__ p, int n) {
    const int t = blockIdx.x * 256 + threadIdx.x;
    if (t < n) p[t] = 0u;
}

// ------------------------------------------------- small FC layers (M = 16)
template <bool HAS_BN, bool RELU>
__global__ __launch_bounds__(256)
void fc_kernel(const float* __restrict__ A, const float* __restrict__ W,
               const float* __restrict__ bias, const float* __restrict__ gamma,
               const float* __restrict__ beta, float* __restrict__ out,
               int K, int N) {
    const int t = blockIdx.x * 256 + threadIdx.x;
    if (t >= B_ * N) return;
    const int row = t / N;
    const int n   = t - row * N;
    const float* ap = A + (size_t)row * K;
    float s = 0.0f;
    for (int k = 0; k < K; ++k) s = fmaf(ap[k], W[(size_t)k * N + n], s);
    s += bias[n];
    if (HAS_BN) s = s * (gamma[n] * rsqrtf(1.0f + 1e-5f)) + beta[n];
    if (RELU)   s = fmaxf(s, 0.0f);
    out[t] = s;
}

// ---------------------------------------------------------------------------
extern "C" void kernel_launch(void* const* d_in, const int* in_sizes, int n_in,
                              void* d_out, int out_size, void* d_ws, size_t ws_size,
                              hipStream_t stream) {
    const float* x     = (const float*)d_in[0];
    const float* W_g1  = (const float*)d_in[1];
    const float* b_g1  = (const float*)d_in[2];
    const float* W_g2  = (const float*)d_in[3];
    const float* b_g2  = (const float*)d_in[4];
    const float* W_l2  = (const float*)d_in[5];
    const float* b_l2  = (const float*)d_in[6];
    const float* W_l3  = (const float*)d_in[7];
    const float* b_l3  = (const float*)d_in[8];
    const float* W_l4  = (const float*)d_in[9];
    const float* b_l4  = (const float*)d_in[10];
    const float* W_l5  = (const float*)d_in[11];
    const float* b_l5  = (const float*)d_in[12];
    const float* W_l6  = (const float*)d_in[13];
    const float* b_l6  = (const float*)d_in[14];
    const float* W_l7  = (const float*)d_in[15];
    const float* b_l7  = (const float*)d_in[16];
    const float* g_bn2 = (const float*)d_in[17];
    const float* be_bn2= (const float*)d_in[18];
    const float* g_bn3 = (const float*)d_in[19];
    const float* be_bn3= (const float*)d_in[20];
    const float* g_bn4 = (const float*)d_in[21];
    const float* be_bn4= (const float*)d_in[22];
    const float* g_bn5 = (const float*)d_in[23];
    const float* be_bn5= (const float*)d_in[24];
    const float* g_bn6 = (const float*)d_in[25];
    const float* be_bn6= (const float*)d_in[26];
    float* out = (float*)d_out;

    // workspace layout
    char* ws = (char*)d_ws;
    int*          idx    = (int*)ws;                               //  4 MiB
    float*        buf1   = (float*)(ws + (size_t)4  * 1024 * 1024);// 64 MiB (<= 32768x512)
    float*        buf2   = (float*)(ws + (size_t)68 * 1024 * 1024);// 64 MiB
    unsigned int* pooled = (unsigned int*)(ws + (size_t)132 * 1024 * 1024); // 64 KiB
    float*        t5     = (float*)(ws + (size_t)132 * 1024 * 1024 + 64 * 1024);
    float*        t6     = (float*)(ws + (size_t)132 * 1024 * 1024 + 128 * 1024);

    // 1) kNN indices (B,N,32)
    knn_kernel<<<NPTS_, 256, 0, stream>>>(x, idx);

    // 2) x @ W_g1 + b  -> buf1 (32768 x 64)
    gemm_g1_kernel<<<(NPTS_ * 64) / 256, 256, 0, stream>>>(x, W_g1, b_g1, buf1);

    // 3) aggregate + relu -> buf2
    agg_relu_kernel<64><<<NPTS_, 64, 0, stream>>>(buf1, idx, buf2);

    // 4) h @ W_g2 + b  (64->64, WMMA) -> buf1
    {
        dim3 grid(NPTS_ / 128, 64 / 64);
        wmma_gemm_kernel<false, false, false><<<grid, 128, 0, stream>>>(
            buf2, W_g2, b_g2, nullptr, nullptr, buf1, nullptr, 64, 64);
    }

    // 5) aggregate + relu -> buf2
    agg_relu_kernel<64><<<NPTS_, 64, 0, stream>>>(buf1, idx, buf2);

    // 6) l2: 64 -> 128, BN2 + relu -> buf1
    {
        dim3 grid(NPTS_ / 128, 128 / 64);
        wmma_gemm_kernel<true, true, false><<<grid, 128, 0, stream>>>(
            buf2, W_l2, b_l2, g_bn2, be_bn2, buf1, nullptr, 64, 128);
    }

    // 7) l3: 128 -> 512, BN3 + relu -> buf2
    {
        dim3 grid(NPTS_ / 128, 512 / 64);
        wmma_gemm_kernel<true, true, false><<<grid, 128, 0, stream>>>(
            buf1, W_l3, b_l3, g_bn3, be_bn3, buf2, nullptr, 128, 512);
    }

    // 8) l4: 512 -> 1024, BN4 + relu, fused max-pool over points -> pooled
    zero_kernel<<<(B_ * 1024 + 255) / 256, 256, 0, stream>>>(pooled, B_ * 1024);
    {
        dim3 grid(NPTS_ / 128, 1024 / 64);
        wmma_gemm_kernel<true, true, true><<<grid, 128, 0, stream>>>(
            buf2, W_l4, b_l4, g_bn4, be_bn4, nullptr, pooled, 512, 1024);
    }

    // 9) l5: 1024 -> 1024, BN5 + relu (pooled bits are valid non-negative floats)
    fc_kernel<true, true><<<(B_ * 1024 + 255) / 256, 256, 0, stream>>>(
        (const float*)pooled, W_l5, b_l5, g_bn5, be_bn5, t5, 1024, 1024);

    // 10) l6: 1024 -> 512, BN6 + relu
    fc_kernel<true, true><<<(B_ * 512 + 255) / 256, 256, 0, stream>>>(
        t5, W_l6, b_l6, g_bn6, be_bn6, t6, 1024, 512);

    // 11) l7: 512 -> 40 -> d_out (16 x 40)
    fc_kernel<false, false><<<(B_ * 40 + 255) / 256, 256, 0, stream>>>(
        t6, W_l7, b_l7, nullptr, nullptr, out, 512, 40);
}